// HiPPOLagT_74131135528970
// MI455X (gfx1250) — compile-verified
//
#include <hip/hip_runtime.h>
#include <hip/hip_bf16.h>

// ---------------------------------------------------------------------------
// HiPPO-LagT scan, N=64, B=64, S=8192, fp32.
// Chunked scan (L=64, 128 chunks) done as:
//   setup:       A_bar, B_bar, GL[j]=A^(63-j)B, MLP[k]=(A^(64*2^k))^T (k=0..6)
//   chunk_local: D_i = F_i @ GL                      (128 WGs, WMMA, parallel)
//   scan x7:     D_i += D_{i-2^k} @ MLP[k]           (Kogge-Stone, parallel)
//   phase2:      per chunk, 64 sequential WMMA steps, stream states to HBM
// All GEMMs use V_WMMA_F32_16X16X4_F32 (fp32 in/out -> no drift over 8192 steps).
//
// Workspace layout (floats):
#define WS_M      0         //  4096  A_bar^T
#define WS_GL     4096      //  4096  GL[j][n] = (A^(63-j) B)[n]
#define WS_BBAR   8192      //    64  B_bar
#define WS_MLP    12288     // 7*4096 (A^(64*2^k))^T, k = 0..6
#define WS_D0     40960     // 128*4096 scan ping
#define WS_D1     565248    // 128*4096 scan pong (final prefix lives here)
// total ~4.2 MB
// ---------------------------------------------------------------------------

typedef __attribute__((ext_vector_type(2))) float v2f;
typedef __attribute__((ext_vector_type(8))) float v8f;

__device__ __forceinline__ v8f wmma_f32(v2f a, v2f b, v8f c) {
  // D(16x16,f32) = A(16x4,f32) x B(4x16,f32) + C
  return __builtin_amdgcn_wmma_f32_16x16x4_f32(
      false, a, false, b, (short)0, c, false, false);
}

// ---------------------------------------------------------------------------
// Phase 0: constants. I-(dt/2)A = diag 1.5 + 0.5*strictly-lower(1);
// I+(dt/2)A = diag 0.5 - 0.5*strictly-lower(1). Forward substitution, then
// 64 matvecs for GL, then 12 squarings (store transposed powers 2^6..2^12).
// ---------------------------------------------------------------------------
__global__ __launch_bounds__(256) void hippo_setup(float* __restrict__ ws) {
  __shared__ __align__(16) float Ab[64 * 64];
  __shared__ __align__(16) float Pm[64 * 64];
  __shared__ __align__(16) float Qm[64 * 64];
  __shared__ float Bb[64];
  __shared__ float Kv[64];
  const int tid = threadIdx.x;

  if (tid < 64) {  // one column of the triangular solve per thread
    const int c = tid;
    float run = 0.f;
    for (int r = 0; r < 64; ++r) {
      const float rhs = (r == c) ? 0.5f : (r > c ? -0.5f : 0.f);
      const float x = (rhs - 0.5f * run) * (1.0f / 1.5f);
      Ab[r * 64 + c] = x;
      run += x;
    }
  }
  if (tid == 64) {  // B_bar
    float run = 0.f;
    for (int r = 0; r < 64; ++r) {
      const float x = (1.0f - 0.5f * run) * (1.0f / 1.5f);
      Bb[r] = x;
      run += x;
    }
  }
  __syncthreads();

  for (int e = tid; e < 4096; e += 256) {  // M = A_bar^T
    const int k = e >> 6, n = e & 63;
    ws[WS_M + e] = Ab[n * 64 + k];
  }
  if (tid < 64) {
    ws[WS_BBAR + tid] = Bb[tid];
    Kv[tid] = Bb[tid];  // K_0 = B_bar
  }
  __syncthreads();

  // GL[63-m][n] = (A^m B)[n]
  for (int m = 0; m < 64; ++m) {
    float nx = 0.f;
    if (tid < 64) {
      ws[WS_GL + (63 - m) * 64 + tid] = Kv[tid];
      for (int j = 0; j < 64; ++j) nx += Ab[tid * 64 + j] * Kv[j];
    }
    __syncthreads();
    if (tid < 64) Kv[tid] = nx;
    __syncthreads();
  }

  // Powers: after s squarings src = A^(2^s); store transposed for s=6..12.
  for (int e = tid; e < 4096; e += 256) Pm[e] = Ab[e];
  __syncthreads();
  float* src = Pm;
  float* dst = Qm;
  for (int s = 1; s <= 12; ++s) {
    for (int e = tid; e < 4096; e += 256) {
      const int r = e >> 6, c = e & 63;
      float sum = 0.f;
      for (int j = 0; j < 64; ++j) sum += src[r * 64 + j] * src[j * 64 + c];
      dst[e] = sum;
    }
    __syncthreads();
    float* t = src; src = dst; dst = t;  // src = A^(2^s)
    if (s >= 6) {
      float* o = ws + WS_MLP + (size_t)(s - 6) * 4096;
      for (int e = tid; e < 4096; e += 256) {
        const int k = e >> 6, n = e & 63;
        o[e] = src[n * 64 + k];  // transposed (right-multiply form)
      }
    }
  }
}

// ---------------------------------------------------------------------------
// D_i = F_i @ GL   (one WG per chunk; wave per 16x16 tile; A streamed from f)
// ---------------------------------------------------------------------------
__global__ __launch_bounds__(512) void hippo_chunk_local(const float* __restrict__ f,
                                                         float* __restrict__ ws) {
  const int ch = blockIdx.x;
  const float* GLp = ws + WS_GL;
  float* D = ws + WS_D0 + (size_t)ch * 4096;

  const int lane = threadIdx.x & 31;
  const int wave = threadIdx.x >> 5;
  const int it = wave >> 2, jt = wave & 3;
  const int lo = lane & 15, hi = lane >> 4;

  v2f bGL[16];
#pragma unroll
  for (int kk = 0; kk < 16; ++kk) {
    const int k0 = 4 * kk + 2 * hi, n = 16 * jt + lo;
    bGL[kk].x = GLp[(k0 + 0) * 64 + n];
    bGL[kk].y = GLp[(k0 + 1) * 64 + n];
  }

  const float* frow = f + (size_t)(16 * it + lo) * 8192 + 64 * ch + 2 * hi;
  v8f acc0 = {}, acc1 = {};  // two chains -> half the WMMA dependency depth
#pragma unroll
  for (int kk = 0; kk < 8; ++kk) {
    v2f a0 = *(const v2f*)&frow[4 * kk];
    acc0 = wmma_f32(a0, bGL[kk], acc0);
    v2f a1 = *(const v2f*)&frow[4 * (kk + 8)];
    acc1 = wmma_f32(a1, bGL[kk + 8], acc1);
  }
#pragma unroll
  for (int r = 0; r < 8; ++r) {
    const int m = 16 * it + r + 8 * hi, n = 16 * jt + lo;
    D[m * 64 + n] = acc0[r] + acc1[r];
  }
}

// ---------------------------------------------------------------------------
// Kogge-Stone scan round k:  D_i <- D_{i-shift} @ MLP[k] + D_i   (i >= shift)
// Uniform P per round => constant matrix; no LDS, no barriers.
// ---------------------------------------------------------------------------
__global__ __launch_bounds__(512) void hippo_scan_step(const float* __restrict__ Dsrc,
                                                       float* __restrict__ Ddst,
                                                       const float* __restrict__ MLPk,
                                                       int shift) {
  const int i = blockIdx.x;
  const int tid = threadIdx.x;
  if (i < shift) {  // uniform per block: plain copy into the pong buffer
    const float* s = Dsrc + (size_t)i * 4096;
    float* d = Ddst + (size_t)i * 4096;
    for (int e = tid; e < 4096; e += 512) d[e] = s[e];
    return;
  }
  const int lane = tid & 31;
  const int wave = tid >> 5;
  const int it = wave >> 2, jt = wave & 3;
  const int lo = lane & 15, hi = lane >> 4;

  v2f bP[16];
#pragma unroll
  for (int kk = 0; kk < 16; ++kk) {
    const int k0 = 4 * kk + 2 * hi, n = 16 * jt + lo;
    bP[kk].x = MLPk[(k0 + 0) * 64 + n];
    bP[kk].y = MLPk[(k0 + 1) * 64 + n];
  }

  const float* Dl = Dsrc + (size_t)(i - shift) * 4096;  // left operand (A)
  const float* Di = Dsrc + (size_t)i * 4096;            // right operand (C)
  float* Do = Ddst + (size_t)i * 4096;

  v8f acc0, acc1 = {};
#pragma unroll
  for (int r = 0; r < 8; ++r)  // C-init = D_i
    acc0[r] = Di[(16 * it + r + 8 * hi) * 64 + 16 * jt + lo];

  const float* arow = Dl + (16 * it + lo) * 64 + 2 * hi;
#pragma unroll
  for (int kk = 0; kk < 8; ++kk) {
    v2f a0 = *(const v2f*)&arow[4 * kk];
    acc0 = wmma_f32(a0, bP[kk], acc0);
    v2f a1 = *(const v2f*)&arow[4 * (kk + 8)];
    acc1 = wmma_f32(a1, bP[kk + 8], acc1);
  }
#pragma unroll
  for (int r = 0; r < 8; ++r) {
    const int m = 16 * it + r + 8 * hi, n = 16 * jt + lo;
    Do[m * 64 + n] = acc0[r] + acc1[r];
  }
}

// ---------------------------------------------------------------------------
// Phase 2: one WG per chunk; 64 sequential steps; state double-buffered in
// LDS (one barrier/step); outputs streamed straight to HBM.
// ---------------------------------------------------------------------------
__global__ __launch_bounds__(512) void hippo_phase2(const float* __restrict__ f,
                                                    const float* __restrict__ ws,
                                                    float* __restrict__ out) {
  __shared__ __align__(16) float cbuf[2][64 * 64];
  __shared__ __align__(16) float fld[64 * 64];  // [batch][t_local]

  const int ch = blockIdx.x;
  const float* Mp = ws + WS_M;
  const float* pre = ws + WS_D1;  // inclusive scan result (7 rounds -> D1)

  const int tid = threadIdx.x;
  const int lane = tid & 31;
  const int wave = tid >> 5;
  const int it = wave >> 2, jt = wave & 3;
  const int lo = lane & 15, hi = lane >> 4;

  const bool first = (ch == 0);
  const float* cs = pre + (size_t)(first ? 0 : (ch - 1)) * 4096;
  for (int e = tid; e < 4096; e += 512) {
    cbuf[0][e] = first ? 0.f : cs[e];  // c_start[ch] = Dscan[ch-1]
    const int b = e >> 6, tp = e & 63;
    fld[e] = f[(size_t)b * 8192 + ch * 64 + tp];
  }

  v2f bM[16];
#pragma unroll
  for (int kk = 0; kk < 16; ++kk) {
    const int k0 = 4 * kk + 2 * hi, n = 16 * jt + lo;
    bM[kk].x = Mp[(k0 + 0) * 64 + n];
    bM[kk].y = Mp[(k0 + 1) * 64 + n];
  }
  const float bb = ws[WS_BBAR + 16 * jt + lo];
  __syncthreads();

  for (int tl = 0; tl < 64; ++tl) {
    const float* cold = cbuf[tl & 1];
    float* cnew = cbuf[(tl + 1) & 1];
    const float* arow = cold + (16 * it + lo) * 64 + 2 * hi;

    v8f acc0 = {}, acc1 = {};
#pragma unroll
    for (int kk = 0; kk < 8; ++kk) {
      v2f a0 = *(const v2f*)&arow[4 * kk];
      acc0 = wmma_f32(a0, bM[kk], acc0);
      v2f a1 = *(const v2f*)&arow[4 * (kk + 8)];
      acc1 = wmma_f32(a1, bM[kk + 8], acc1);
    }

    const size_t t = (size_t)ch * 64 + tl;
#pragma unroll
    for (int r = 0; r < 8; ++r) {
      const int m = 16 * it + r + 8 * hi;
      const int n = 16 * jt + lo;
      const float v = acc0[r] + acc1[r] + fld[m * 64 + tl] * bb;
      cnew[m * 64 + n] = v;
      out[(size_t)m * 8192 * 64 + t * 64 + n] = v;
    }
    __syncthreads();
  }
}

extern "C" void kernel_launch(void* const* d_in, const int* in_sizes, int n_in,
                              void* d_out, int out_size, void* d_ws, size_t ws_size,
                              hipStream_t stream) {
  (void)in_sizes; (void)n_in; (void)out_size; (void)ws_size;
  const float* f = (const float*)d_in[0];  // (64, 8192) fp32
  float* out = (float*)d_out;              // (64, 8192, 64) fp32
  float* ws = (float*)d_ws;                // needs ~4.2 MB

  hippo_setup<<<1, 256, 0, stream>>>(ws);
  hippo_chunk_local<<<128, 512, 0, stream>>>(f, ws);

  float* D0 = ws + WS_D0;
  float* D1 = ws + WS_D1;
  for (int k = 0; k < 7; ++k) {  // 7 rounds: D0->D1->D0->...->D1 (final = D1)
    const float* src = (k & 1) ? D1 : D0;
    float* dst = (k & 1) ? D0 : D1;
    hippo_scan_step<<<128, 512, 0, stream>>>(src, dst, ws + WS_MLP + (size_t)k * 4096,
                                             1 << k);
  }

  hippo_phase2<<<128, 512, 0, stream>>>(f, ws, out);
}